// AdditiveAttention_4569845203238
// MI455X (gfx1250) — compile-verified
//
#include <hip/hip_runtime.h>
#include <math.h>

typedef float v2f __attribute__((ext_vector_type(2)));
typedef float v8f __attribute__((ext_vector_type(8)));
typedef int   v4i __attribute__((ext_vector_type(4)));

#define HDIM 256
#define QLEN 512
#define KLEN 512
#define NBATCH 4

// ---------------------------------------------------------------------------
// Hardware tanh: CDNA5 has single-instruction transcendental v_tanh_f32.
// The score phase is transcendental-VALU bound, so this is the key lever.
// ---------------------------------------------------------------------------
__device__ __forceinline__ float htanh(float x) {
#if __has_builtin(__builtin_amdgcn_tanhf)
    return __builtin_amdgcn_tanhf(x);
#else
    float r;
    // TRANS op: 1 independent op required before result use (ISA 7.4) -> v_nop
    asm("v_tanh_f32 %0, %1\n\tv_nop" : "=v"(r) : "v"(x));
    return r;
#endif
}

// ---------------------------------------------------------------------------
// Async global -> LDS copy (16B per lane), tracked by ASYNCcnt (ISA ch.8 §4).
// Probe feedback: param 0 is `int __vector(4) *` (non-const). Cast accordingly.
// Falls back to a synchronous float4 copy if the builtin is not declared.
// ---------------------------------------------------------------------------
#if __has_builtin(__builtin_amdgcn_global_load_async_to_lds_b128)
__device__ __forceinline__ void async_copy16(const float* g, float* l) {
    __builtin_amdgcn_global_load_async_to_lds_b128((v4i*)g, (v4i*)l, 0, 0);
}
#else
__device__ __forceinline__ void async_copy16(const float* g, float* l) {
    *(float4*)l = *(const float4*)g;
}
#endif

__device__ __forceinline__ void wait_async0() {
#if __has_builtin(__builtin_amdgcn_s_wait_asynccnt)
    __builtin_amdgcn_s_wait_asynccnt(0);
#else
    asm volatile("s_wait_asynccnt 0" ::: "memory");
#endif
}

// ---------------------------------------------------------------------------
// Projection GEMM: Y[m][o] = sum_h X[m][h] * W[o][h]   (X @ W^T)
// One wave per 16x16 output tile, V_WMMA_F32_16X16X4_F32, K-loop of 64.
// ---------------------------------------------------------------------------
__global__ __launch_bounds__(128) void proj_gemm_wmma(
    const float* __restrict__ X, const float* __restrict__ W,
    float* __restrict__ Y, int M)
{
    const int lane   = threadIdx.x & 31;
    const int wave   = threadIdx.x >> 5;
    const int waveId = blockIdx.x * 4 + wave;
    const int tilesN = HDIM / 16;                 // 16
    const int tm     = (waveId / tilesN) * 16;    // M tile base
    const int tn     = (waveId % tilesN) * 16;    // N tile base
    if (tm >= M) return;

    const int row   = lane & 15;                  // M (for A) / N (for B)
    const int khalf = (lane >> 4) * 2;            // lanes 16-31 hold K+2,K+3

    const float* xrow = X + (size_t)(tm + row) * HDIM; // A: X[tm+row][*]
    const float* wrow = W + (size_t)(tn + row) * HDIM; // B[k][n] = W[n][k]

    v8f acc = {};
    for (int k0 = 0; k0 < HDIM; k0 += 4) {
        v2f a = *(const v2f*)(xrow + k0 + khalf);  // (K, K+1) this half-wave
        v2f b = *(const v2f*)(wrow + k0 + khalf);  // W[n][K], W[n][K+1]
        acc = __builtin_amdgcn_wmma_f32_16x16x4_f32(
            false, a, false, b, (short)0, acc, false, false);
    }

    #pragma unroll
    for (int r = 0; r < 8; ++r) {
        int m = tm + r + (lane >> 4) * 8;
        int n = tn + (lane & 15);
        Y[(size_t)m * HDIM + n] = acc[r];
    }
}

// ---------------------------------------------------------------------------
// Fused additive-attention core. One block = (batch, 16 query rows).
// Phase B: v_tanh scores; k-tiles double-buffered via async global->LDS.
// Phase C: row softmax (v_exp).
// Phase D: attn @ V with V_WMMA_F32_16X16X4_F32, A fragments from LDS.
// ---------------------------------------------------------------------------
#define KSTRIDE 260   // row stride: %64==4 (conflict-free) and 16B-aligned

__global__ __launch_bounds__(256) void fused_additive_attn(
    const float* __restrict__ qp,   // [B*Q, H]
    const float* __restrict__ kp,   // [B*K, H]
    const float* __restrict__ V,    // [B, K, H]
    const float* __restrict__ wv,   // [H]
    float* __restrict__ out)        // [B, Q, H]
{
    __shared__ float qt[16][KSTRIDE];
    __shared__ float kt[2][16][KSTRIDE];   // double buffer for async stream
    __shared__ float s [16][513];          // scores -> attn (stride%64==1)
    __shared__ float wvs[HDIM];
    __shared__ float red[16][16];
    __shared__ float rowmax[16];
    __shared__ float rowinv[16];

    const int tid = threadIdx.x;
    const int b   = blockIdx.x >> 5;           // 32 q-tiles per batch
    const int q0  = (blockIdx.x & 31) * 16;

    wvs[tid] = wv[tid];
    // q tile: 16 rows x 256 floats = 1024 float4, 4 per thread
    {
        const float* qbase = qp + (size_t)(b * QLEN + q0) * HDIM;
        #pragma unroll
        for (int j = 0; j < 4; ++j) {
            int i = tid + j * 256;
            int r = i >> 6, c = (i & 63) * 4;
            *(float4*)&qt[r][c] = *(const float4*)(qbase + (size_t)r * HDIM + c);
        }
    }

    // prologue: async-stream k tile 0 into buffer 0
    const float* kbase = kp + (size_t)(b * KLEN) * HDIM;
    #pragma unroll
    for (int j = 0; j < 4; ++j) {
        int i = tid + j * 256;
        int r = i >> 6, c = (i & 63) * 4;
        async_copy16(kbase + (size_t)r * HDIM + c, &kt[0][r][c]);
    }
    wait_async0();
    __syncthreads();

    // ---- Phase B: scores[qi][k] = sum_h wv[h]*tanh(qt+kt), 1 (q,k) pair/thread
    const int qi = tid >> 4;
    const int kj = tid & 15;
    for (int ktile = 0; ktile < KLEN / 16; ++ktile) {
        const int cur = ktile & 1;
        if (ktile + 1 < KLEN / 16) {       // prefetch next tile into other buf
            const float* nbase = kbase + (size_t)(ktile + 1) * 16 * HDIM;
            #pragma unroll
            for (int j = 0; j < 4; ++j) {
                int i = tid + j * 256;
                int r = i >> 6, c = (i & 63) * 4;
                async_copy16(nbase + (size_t)r * HDIM + c, &kt[cur ^ 1][r][c]);
            }
        }

        const float4* qv = (const float4*)&qt[qi][0];
        const float4* kv = (const float4*)&kt[cur][kj][0];
        const float4* wq = (const float4*)wvs;
        float accv = 0.f;
        #pragma unroll 2
        for (int h4 = 0; h4 < HDIM / 4; ++h4) {
            float4 qq = qv[h4], kk = kv[h4], ww = wq[h4];
            float t0 = htanh(qq.x + kk.x);
            float t1 = htanh(qq.y + kk.y);
            float t2 = htanh(qq.z + kk.z);
            float t3 = htanh(qq.w + kk.w);
            accv = fmaf(ww.x, t0, accv);
            accv = fmaf(ww.y, t1, accv);
            accv = fmaf(ww.z, t2, accv);
            accv = fmaf(ww.w, t3, accv);
        }
        s[qi][ktile * 16 + kj] = accv;

        wait_async0();      // next tile landed in LDS
        __syncthreads();    // visible to all waves; s writes also fenced
    }

    // ---- Phase C: softmax over K=512 per row; 16 threads/row, 32 cols each
    {
        const int row  = tid >> 4;
        const int part = tid & 15;
        float m = -1e30f;
        for (int c = part * 32; c < part * 32 + 32; ++c) m = fmaxf(m, s[row][c]);
        red[row][part] = m;
        __syncthreads();
        if (part == 0) {
            float mm = red[row][0];
            #pragma unroll
            for (int i = 1; i < 16; ++i) mm = fmaxf(mm, red[row][i]);
            rowmax[row] = mm;
        }
        __syncthreads();
        const float mm = rowmax[row];
        float sum = 0.f;
        for (int c = part * 32; c < part * 32 + 32; ++c) {
            float e = __expf(s[row][c] - mm);                // v_exp_f32
            s[row][c] = e;
            sum += e;
        }
        red[row][part] = sum;
        __syncthreads();
        if (part == 0) {
            float ss = 0.f;
            #pragma unroll
            for (int i = 0; i < 16; ++i) ss += red[row][i];
            rowinv[row] = 1.0f / ss;
        }
        __syncthreads();
        const float inv = rowinv[row];
        for (int c = part * 32; c < part * 32 + 32; ++c) s[row][c] *= inv;
        __syncthreads();
    }

    // ---- Phase D: O[16][256] = attn[16][512] @ V[b]  via fp32 WMMA
    const int lane  = tid & 31;
    const int wave  = tid >> 5;                 // 8 waves, 2 N-tiles each
    const int mrow  = lane & 15;
    const int khalf = (lane >> 4) * 2;
    for (int nt = wave; nt < 16; nt += 8) {
        const int n0 = nt * 16;
        v8f acc = {};
        for (int k0 = 0; k0 < KLEN; k0 += 4) {
            v2f a;                                  // A from LDS attn matrix
            a.x = s[mrow][k0 + khalf];
            a.y = s[mrow][k0 + khalf + 1];
            const float* vp = V + ((size_t)b * KLEN + k0 + khalf) * HDIM + n0 + mrow;
            v2f bb;                                 // B[k][n] = V[b][k][n]
            bb.x = vp[0];
            bb.y = vp[HDIM];
            acc = __builtin_amdgcn_wmma_f32_16x16x4_f32(
                false, a, false, bb, (short)0, acc, false, false);
        }
        #pragma unroll
        for (int r = 0; r < 8; ++r) {
            int m = r + (lane >> 4) * 8;
            out[((size_t)b * QLEN + q0 + m) * HDIM + n0 + (lane & 15)] = acc[r];
        }
    }
}

// ---------------------------------------------------------------------------
extern "C" void kernel_launch(void* const* d_in, const int* in_sizes, int n_in,
                              void* d_out, int out_size, void* d_ws, size_t ws_size,
                              hipStream_t stream) {
    const float* queries = (const float*)d_in[0];  // [4,512,256]
    const float* keys    = (const float*)d_in[1];  // [4,512,256]
    const float* values  = (const float*)d_in[2];  // [4,512,256]
    const float* W_q     = (const float*)d_in[3];  // [256,256]
    const float* W_k     = (const float*)d_in[4];  // [256,256]
    const float* w_v     = (const float*)d_in[5];  // [256]
    float* out = (float*)d_out;

    const int M = NBATCH * QLEN;                   // 2048 rows
    float* qp = (float*)d_ws;                      // [2048,256] = 2 MB
    float* kp = qp + (size_t)M * HDIM;             // [2048,256] = 2 MB

    const int nTiles = (M / 16) * (HDIM / 16);
    dim3 gridProj(nTiles / 4);
    proj_gemm_wmma<<<gridProj, 128, 0, stream>>>(queries, W_q, qp, M);
    proj_gemm_wmma<<<gridProj, 128, 0, stream>>>(keys,    W_k, kp, M);

    dim3 gridAttn(NBATCH * (QLEN / 16));
    fused_additive_attn<<<gridAttn, 256, 0, stream>>>(qp, kp, values, w_v, out);
}